// VladPooling_42640435314929
// MI455X (gfx1250) — compile-verified
//
#include <hip/hip_runtime.h>

#define KCENT 10
#define KC    12
#define NB    16
#define NS    1024
#define ND    512

typedef __attribute__((ext_vector_type(2))) float v2f;
typedef __attribute__((ext_vector_type(4))) float v4f;
typedef __attribute__((ext_vector_type(8))) float v8f;

// ---------------- Phase 1: softmax + transposed store of A^T (padded to 16 rows) ----
__global__ __launch_bounds__(256) void vlad_softmax_kernel(
    const float* __restrict__ cluster_score,  // (B, S, KC)
    float* __restrict__ At,                   // (B, 16, S)  rows 12..15 zeroed
    float* __restrict__ sq)                   // (B*KCENT) accumulators -> zero
{
    const int gid = blockIdx.x * 256 + threadIdx.x;   // b*S + s, exactly 16384 threads
    if (blockIdx.x == 0 && threadIdx.x < NB * KCENT) sq[threadIdx.x] = 0.0f;

    const int b = gid >> 10;
    const int s = gid & (NS - 1);

    const float* src = cluster_score + (size_t)gid * KC;   // 48B stride -> 16B aligned
    float v[KC];
    v4f x0 = *(const v4f*)(src + 0);
    v4f x1 = *(const v4f*)(src + 4);
    v4f x2 = *(const v4f*)(src + 8);
    v[0]=x0.x; v[1]=x0.y; v[2]=x0.z; v[3]=x0.w;
    v[4]=x1.x; v[5]=x1.y; v[6]=x1.z; v[7]=x1.w;
    v[8]=x2.x; v[9]=x2.y; v[10]=x2.z; v[11]=x2.w;

    float m = v[0];
#pragma unroll
    for (int k = 1; k < KC; ++k) m = fmaxf(m, v[k]);
    float sum = 0.0f;
#pragma unroll
    for (int k = 0; k < KC; ++k) { v[k] = __expf(v[k] - m); sum += v[k]; }
    const float inv = 1.0f / sum;

    float* dst = At + ((size_t)b * 16) * NS + s;  // column s, stride NS between rows
#pragma unroll
    for (int k = 0; k < KC; ++k) dst[(size_t)k * NS] = v[k] * inv;
#pragma unroll
    for (int k = KC; k < 16; ++k) dst[(size_t)k * NS] = 0.0f;
}

// ---------------- Phase 2: WMMA f32 16x16x4 GEMM + colsum correction + sq reduce ----
// grid.x = NB * 8 (8 d-groups of 64 cols), block = 128 (4 waves, one 16-col tile each)
__global__ __launch_bounds__(128) void vlad_gemm_kernel(
    const float* __restrict__ feat,     // (B, S, D)
    const float* __restrict__ cluster,  // (KC, D)
    const float* __restrict__ At,       // (B, 16, S)
    float* __restrict__ res,            // (B, KCENT, D)
    float* __restrict__ sq)             // (B*KCENT)
{
    const int b    = blockIdx.x >> 3;
    const int dgrp = blockIdx.x & 7;
    const int wave = threadIdx.x >> 5;
    const int lane = threadIdx.x & 31;
    const int half = lane >> 4;       // 0: rows 0-7 of C, 1: rows 8-15
    const int l    = lane & 15;
    const int d    = dgrp * 64 + wave * 16 + l;

    // A-frag: lane l (half h) supplies row M=l, K = s0+2h, s0+2h+1 (ISA 16x4 f32 layout)
    const float* Ab = At   + ((size_t)b * 16 + l) * NS;
    const float* Fb = feat + (size_t)b * NS * ND + d;

    v8f c0 = {}, c1 = {}, c2 = {}, c3 = {};
    float sumA = 0.0f;

    for (int s0 = 0; s0 < NS; s0 += 16) {
        // prefetch the feat stream ahead, WGP scope (pull through all cache levels)
        __builtin_prefetch(Fb + (size_t)(s0 + 64) * ND, 0, 3);

        // ---- chunk 0: K = s0 .. s0+3 ----
        v2f a0 = *(const v2f*)(Ab + s0 + 2 * half);
        v2f b0;
        b0.x = __builtin_nontemporal_load(&Fb[(size_t)(s0 + 2 * half + 0) * ND]);
        b0.y = __builtin_nontemporal_load(&Fb[(size_t)(s0 + 2 * half + 1) * ND]);
        c0 = __builtin_amdgcn_wmma_f32_16x16x4_f32(false, a0, false, b0,
                                                   (short)0, c0, false, false);
        // ---- chunk 1: K = s0+4 .. s0+7 ----
        v2f a1 = *(const v2f*)(Ab + s0 + 4 + 2 * half);
        v2f b1;
        b1.x = __builtin_nontemporal_load(&Fb[(size_t)(s0 + 4 + 2 * half + 0) * ND]);
        b1.y = __builtin_nontemporal_load(&Fb[(size_t)(s0 + 4 + 2 * half + 1) * ND]);
        c1 = __builtin_amdgcn_wmma_f32_16x16x4_f32(false, a1, false, b1,
                                                   (short)0, c1, false, false);
        // ---- chunk 2: K = s0+8 .. s0+11 ----
        v2f a2 = *(const v2f*)(Ab + s0 + 8 + 2 * half);
        v2f b2;
        b2.x = __builtin_nontemporal_load(&Fb[(size_t)(s0 + 8 + 2 * half + 0) * ND]);
        b2.y = __builtin_nontemporal_load(&Fb[(size_t)(s0 + 8 + 2 * half + 1) * ND]);
        c2 = __builtin_amdgcn_wmma_f32_16x16x4_f32(false, a2, false, b2,
                                                   (short)0, c2, false, false);
        // ---- chunk 3: K = s0+12 .. s0+15 ----
        v2f a3 = *(const v2f*)(Ab + s0 + 12 + 2 * half);
        v2f b3;
        b3.x = __builtin_nontemporal_load(&Fb[(size_t)(s0 + 12 + 2 * half + 0) * ND]);
        b3.y = __builtin_nontemporal_load(&Fb[(size_t)(s0 + 12 + 2 * half + 1) * ND]);
        c3 = __builtin_amdgcn_wmma_f32_16x16x4_f32(false, a3, false, b3,
                                                   (short)0, c3, false, false);

        sumA += (a0.x + a0.y) + (a1.x + a1.y) + (a2.x + a2.y) + (a3.x + a3.y);
    }

    float c[8];
#pragma unroll
    for (int i = 0; i < 8; ++i) c[i] = (c0[i] + c1[i]) + (c2[i] + c3[i]);

    // colsum[k]: lane j holds partial for k=j&15 over its half of s; fold halves.
    const float tot = sumA + __shfl_xor(sumA, 16, 32);  // lane j -> colsum[j & 15]

#pragma unroll
    for (int m = 0; m < 8; ++m) {
        const int k = half * 8 + m;                 // C row this VGPR holds
        if (k < KCENT) {                            // uniform per 16-lane half
            const float cs = __shfl(tot, k, 32);
            const float r  = c[m] - cs * cluster[(size_t)k * ND + d];
            res[((size_t)b * KCENT + k) * ND + d] = r;
            float s2 = r * r;                       // reduce over 16 cols (within half)
            s2 += __shfl_xor(s2, 1, 32);
            s2 += __shfl_xor(s2, 2, 32);
            s2 += __shfl_xor(s2, 4, 32);
            s2 += __shfl_xor(s2, 8, 32);
            if (l == 0) atomicAdd(&sq[b * KCENT + k], s2);
        }
    }
}

// ---------------- Phase 3: L2 normalize -----------------------------------------
__global__ __launch_bounds__(256) void vlad_norm_kernel(
    const float* __restrict__ res, const float* __restrict__ sq,
    float* __restrict__ out)
{
    const int i  = blockIdx.x * 256 + threadIdx.x;    // 81920 total
    const int bk = i >> 9;                            // b*KCENT + k
    out[i] = res[i] * rsqrtf(fmaxf(sq[bk], 1e-12f));
}

extern "C" void kernel_launch(void* const* d_in, const int* in_sizes, int n_in,
                              void* d_out, int out_size, void* d_ws, size_t ws_size,
                              hipStream_t stream) {
    const float* feat          = (const float*)d_in[0];
    const float* cluster_score = (const float*)d_in[1];
    const float* cluster       = (const float*)d_in[2];
    float* out = (float*)d_out;

    char* ws = (char*)d_ws;
    float* At  = (float*)ws;                                        // 16*16*1024 f32 = 1 MB
    float* res = (float*)(ws + (size_t)NB * 16 * NS * 4);           // 16*10*512 f32 = 320 KB
    float* sq  = (float*)(ws + (size_t)NB * 16 * NS * 4
                             + (size_t)NB * KCENT * ND * 4);        // 160 f32

    vlad_softmax_kernel<<<64, 256, 0, stream>>>(cluster_score, At, sq);
    vlad_gemm_kernel<<<NB * 8, 128, 0, stream>>>(feat, cluster, At, res, sq);
    vlad_norm_kernel<<<(NB * KCENT * ND) / 256, 256, 0, stream>>>(res, sq, out);
}